// GAT2_1709396984305
// MI455X (gfx1250) — compile-verified
//
#include <hip/hip_runtime.h>

typedef __attribute__((ext_vector_type(2))) float v2f;
typedef __attribute__((ext_vector_type(8))) float v8f;

#define N_NODES 100000
#define N_EDGES 1600000
#define ETOT    (N_EDGES + N_NODES)
#define FDIM    32
#define NEG_SLOPE 0.2f

// ---------------- helpers ----------------
__device__ __forceinline__ unsigned fkey(float f) {
  unsigned u = __float_as_uint(f);
  return (u & 0x80000000u) ? ~u : (u | 0x80000000u);
}
__device__ __forceinline__ float fkey_inv(unsigned k) {
  return (k & 0x80000000u) ? __uint_as_float(k ^ 0x80000000u) : __uint_as_float(~k);
}

// ------------- self-loop 'mean' fill -------------
__global__ void k_degree(const int* __restrict__ dst, const float* __restrict__ ea,
                         float* __restrict__ cnt, float* __restrict__ meanattr) {
  int e = blockIdx.x * blockDim.x + threadIdx.x;
  if (e >= N_EDGES) return;
  int d = dst[e];
  unsafeAtomicAdd(&cnt[d], 1.0f);
  unsafeAtomicAdd(&meanattr[d], ea[e]);
}
__global__ void k_mean(float* __restrict__ meanattr, const float* __restrict__ cnt) {
  int i = blockIdx.x * blockDim.x + threadIdx.x;
  if (i >= N_NODES) return;
  meanattr[i] = meanattr[i] / fmaxf(cnt[i], 1.0f);
}

// ------------- WMMA GEMM: XL = X@Wl + bl ; XR = X@Wr + br -------------
// One wave computes a 16x32 tile of both outputs. FIN = 128 or 32, FOUT = 32.
template <int FIN>
__global__ void k_gemm_lr(const float* __restrict__ X,
                          const float* __restrict__ Wl, const float* __restrict__ bl,
                          const float* __restrict__ Wr, const float* __restrict__ br,
                          float* __restrict__ XL, float* __restrict__ XR, int ntiles) {
  int wave = (int)((blockIdx.x * blockDim.x + threadIdx.x) >> 5);
  if (wave >= ntiles) return;                 // wave-uniform guard: EXEC stays all-ones
  int lane = threadIdx.x & 31;
  int hi = lane >> 4;                         // 0: lanes 0-15, 1: lanes 16-31
  int lo = lane & 15;
  int arow = wave * 16 + lo;                  // A-matrix row for this lane

  v8f cl0 = {}, cl1 = {}, cr0 = {}, cr1 = {};
  for (int k0 = 0; k0 < FIN; k0 += 4) {
    int ka = k0 + hi * 2;                     // ISA A layout: v0={K0,K2}, v1={K1,K3}
    v2f a;
    a.x = X[arow * FIN + ka + 0];
    a.y = X[arow * FIN + ka + 1];
    v2f b0, b1, b2, b3;                       // B layout mirrored: rows ka, ka+1
    b0.x = Wl[(ka + 0) * FDIM + lo];       b0.y = Wl[(ka + 1) * FDIM + lo];
    b1.x = Wl[(ka + 0) * FDIM + lo + 16];  b1.y = Wl[(ka + 1) * FDIM + lo + 16];
    b2.x = Wr[(ka + 0) * FDIM + lo];       b2.y = Wr[(ka + 1) * FDIM + lo];
    b3.x = Wr[(ka + 0) * FDIM + lo + 16];  b3.y = Wr[(ka + 1) * FDIM + lo + 16];
    cl0 = __builtin_amdgcn_wmma_f32_16x16x4_f32(false, a, false, b0, (short)0, cl0, false, false);
    cl1 = __builtin_amdgcn_wmma_f32_16x16x4_f32(false, a, false, b1, (short)0, cl1, false, false);
    cr0 = __builtin_amdgcn_wmma_f32_16x16x4_f32(false, a, false, b2, (short)0, cr0, false, false);
    cr1 = __builtin_amdgcn_wmma_f32_16x16x4_f32(false, a, false, b3, (short)0, cr1, false, false);
  }
  // C/D layout: VGPR r -> (M=r, N=lane) lanes 0-15 ; (M=r+8, N=lane-16) lanes 16-31
  #pragma unroll
  for (int r = 0; r < 8; ++r) {
    int orow = wave * 16 + r + hi * 8;
    XL[orow * FDIM + lo]      = cl0[r] + bl[lo];
    XL[orow * FDIM + lo + 16] = cl1[r] + bl[lo + 16];
    XR[orow * FDIM + lo]      = cr0[r] + br[lo];
    XR[orow * FDIM + lo + 16] = cr1[r] + br[lo + 16];
  }
}

// ------------- edge pass 1: score + segment max -------------
// one wave per edge; lane = feature; coalesced 128B row gathers
__global__ void k_edge_score(const float* __restrict__ XL, const float* __restrict__ XR,
                             const int* __restrict__ src, const int* __restrict__ dst,
                             const float* __restrict__ ea, const float* __restrict__ meanattr,
                             const float* __restrict__ We, const float* __restrict__ att,
                             float* __restrict__ sbuf, unsigned* __restrict__ maxkey) {
  int wave = (int)((blockIdx.x * blockDim.x + threadIdx.x) >> 5);
  if (wave >= ETOT) return;
  int lane = threadIdx.x & 31;
  int s, d; float eav;
  if (wave < N_EDGES) { s = src[wave]; d = dst[wave]; eav = ea[wave]; }
  else { s = wave - N_EDGES; d = s; eav = meanattr[s]; }
  float v = XL[s * FDIM + lane] + XR[d * FDIM + lane] + eav * We[lane];
  v = v > 0.0f ? v : NEG_SLOPE * v;
  float t = v * att[lane];
  #pragma unroll
  for (int off = 16; off; off >>= 1) t += __shfl_xor(t, off, 32);
  if (lane == 0) {
    sbuf[wave] = t;
    atomicMax(&maxkey[d], fkey(t));
  }
}

// ------------- edge pass 2: p = exp(s - m), z += p -------------
__global__ void k_edge_exp(const int* __restrict__ dst, float* __restrict__ sbuf,
                           const unsigned* __restrict__ maxkey, float* __restrict__ z) {
  int e = blockIdx.x * blockDim.x + threadIdx.x;
  if (e >= ETOT) return;
  int d = (e < N_EDGES) ? dst[e] : (e - N_EDGES);
  float m = fkey_inv(maxkey[d]);
  float p = expf(sbuf[e] - m);
  sbuf[e] = p;
  unsafeAtomicAdd(&z[d], p);
}

// ------------- edge pass 3: out[dst] += alpha * xl[src] -------------
__global__ void k_edge_aggr(const float* __restrict__ XL,
                            const int* __restrict__ src, const int* __restrict__ dst,
                            const float* __restrict__ sbuf, const float* __restrict__ z,
                            float* __restrict__ out) {
  int wave = (int)((blockIdx.x * blockDim.x + threadIdx.x) >> 5);
  if (wave >= ETOT) return;
  int lane = threadIdx.x & 31;
  int s, d;
  if (wave < N_EDGES) { s = src[wave]; d = dst[wave]; }
  else { s = wave - N_EDGES; d = s; }
  float alpha = sbuf[wave] / z[d];
  unsafeAtomicAdd(&out[d * FDIM + lane], alpha * XL[s * FDIM + lane]);
}

// ------------- per-node finalize -------------
__global__ void k_bias_relu(float* __restrict__ C, const float* __restrict__ b) {
  int i = blockIdx.x * blockDim.x + threadIdx.x;
  if (i >= N_NODES * FDIM) return;
  C[i] = fmaxf(C[i] + b[i & 31], 0.0f);
}
__global__ void k_bias(float* __restrict__ C, const float* __restrict__ b) {
  int i = blockIdx.x * blockDim.x + threadIdx.x;
  if (i >= N_NODES * FDIM) return;
  C[i] = C[i] + b[i & 31];
}

extern "C" void kernel_launch(void* const* d_in, const int* in_sizes, int n_in,
                              void* d_out, int out_size, void* d_ws, size_t ws_size,
                              hipStream_t stream) {
  const float* x    = (const float*)d_in[0];
  const int*   src  = (const int*)d_in[1];
  const int*   dst  = (const int*)d_in[2];
  const float* ea   = (const float*)d_in[3];
  const float* Wl1  = (const float*)d_in[4];
  const float* bl1  = (const float*)d_in[5];
  const float* Wr1  = (const float*)d_in[6];
  const float* br1  = (const float*)d_in[7];
  const float* We1  = (const float*)d_in[8];
  const float* att1 = (const float*)d_in[9];
  const float* b1   = (const float*)d_in[10];
  const float* Wl2  = (const float*)d_in[11];
  const float* bl2  = (const float*)d_in[12];
  const float* Wr2  = (const float*)d_in[13];
  const float* br2  = (const float*)d_in[14];
  const float* We2  = (const float*)d_in[15];
  const float* att2 = (const float*)d_in[16];
  const float* b2   = (const float*)d_in[17];
  float* out = (float*)d_out;

  // workspace layout
  float*    A        = (float*)d_ws;           // xl   [N,32]
  float*    B        = A + (size_t)N_NODES * FDIM;   // xr   [N,32]
  float*    C        = B + (size_t)N_NODES * FDIM;   // layer1 accum -> h
  float*    sbuf     = C + (size_t)N_NODES * FDIM;   // scores/p [ETOT]
  unsigned* maxkey   = (unsigned*)(sbuf + ETOT);     // [N]
  float*    z        = (float*)(maxkey + N_NODES);   // [N]
  float*    cnt      = z + N_NODES;                  // [N]
  float*    meanattr = cnt + N_NODES;                // [N]

  const int TB = 256;
  const int ntiles = N_NODES / 16;                       // 6250
  dim3 blk(TB);
  dim3 g_gemm((ntiles * 32 + TB - 1) / TB);              // 1 wave / 16-row tile
  dim3 g_edgeW((ETOT * 32 + TB - 1) / TB);               // 1 wave / edge
  dim3 g_edgeT((ETOT + TB - 1) / TB);                    // 1 thread / edge
  dim3 g_E((N_EDGES + TB - 1) / TB);
  dim3 g_N((N_NODES + TB - 1) / TB);
  dim3 g_NF((N_NODES * FDIM + TB - 1) / TB);

  // zero maxkey,z,cnt,meanattr (contiguous 4*N words); key(x) > 0 for all x, so 0 == -inf
  hipMemsetAsync(maxkey, 0, (size_t)4 * N_NODES * 4, stream);
  k_degree<<<g_E, blk, 0, stream>>>(dst, ea, cnt, meanattr);
  k_mean<<<g_N, blk, 0, stream>>>(meanattr, cnt);

  // ---------- layer 1 ----------
  k_gemm_lr<128><<<g_gemm, blk, 0, stream>>>(x, Wl1, bl1, Wr1, br1, A, B, ntiles);
  hipMemsetAsync(C, 0, (size_t)N_NODES * FDIM * 4, stream);
  k_edge_score<<<g_edgeW, blk, 0, stream>>>(A, B, src, dst, ea, meanattr, We1, att1, sbuf, maxkey);
  k_edge_exp<<<g_edgeT, blk, 0, stream>>>(dst, sbuf, maxkey, z);
  k_edge_aggr<<<g_edgeW, blk, 0, stream>>>(A, src, dst, sbuf, z, C);
  k_bias_relu<<<g_NF, blk, 0, stream>>>(C, b1);      // C = h = relu(out1 + b1)

  // ---------- layer 2 ----------
  hipMemsetAsync(maxkey, 0, (size_t)2 * N_NODES * 4, stream);  // maxkey + z
  hipMemsetAsync(out, 0, (size_t)N_NODES * FDIM * 4, stream);
  k_gemm_lr<32><<<g_gemm, blk, 0, stream>>>(C, Wl2, bl2, Wr2, br2, A, B, ntiles);
  k_edge_score<<<g_edgeW, blk, 0, stream>>>(A, B, src, dst, ea, meanattr, We2, att2, sbuf, maxkey);
  k_edge_exp<<<g_edgeT, blk, 0, stream>>>(dst, sbuf, maxkey, z);
  k_edge_aggr<<<g_edgeW, blk, 0, stream>>>(A, src, dst, sbuf, z, out);
  k_bias<<<g_NF, blk, 0, stream>>>(out, b2);
}